// GraphConvolution_27513560498274
// MI455X (gfx1250) — compile-verified
//
#include <hip/hip_runtime.h>
#include <hip/hip_bf16.h>

#define N_NODES 100000
#define D_IN    256
#define D_OUT   64

typedef __attribute__((ext_vector_type(2))) float v2f;
typedef __attribute__((ext_vector_type(8))) float v8f;

// ---------------------------------------------------------------------------
// Kernel 1: support = x @ W using V_WMMA_F32_16X16X4_F32 (fp32 matrix path).
// One block = 128 threads = 4 wave32s; block handles a 16-row strip of x,
// wave w handles output column tile [w*16, w*16+16). 6250 blocks cover all
// 100000 rows exactly (100000 = 16 * 6250), D_OUT = 64 = 4 * 16 exactly.
// ---------------------------------------------------------------------------
__global__ __launch_bounds__(128) void gcn_gemm_wmma(
    const float* __restrict__ x,      // [N_NODES, D_IN]
    const float* __restrict__ wgt,    // [D_IN, D_OUT]
    float* __restrict__ support) {    // [N_NODES, D_OUT]
  const int lane  = threadIdx.x & 31;
  const int wave  = threadIdx.x >> 5;          // n-tile 0..3
  const int row0  = blockIdx.x * 16;           // m-tile base row
  const int m     = lane & 15;                 // A row in tile / B,C column in tile
  const int khalf = (lane >> 4) << 1;          // 0 for lanes 0-15, 2 for lanes 16-31
  const int ncol  = wave * 16 + m;             // global output column

  // A: lane holds x[row0+m, k+khalf] and x[row0+m, k+khalf+1] (consecutive -> b64 load)
  const float* arow = x + (size_t)(row0 + m) * D_IN + khalf;
  // B: lane holds wgt[k+khalf, ncol] and wgt[k+khalf+1, ncol]
  const float* bcol = wgt + (size_t)khalf * D_OUT + ncol;

  v8f c = {};
#pragma unroll 4
  for (int k = 0; k < D_IN; k += 4) {
    v2f a, b;
    a.x = arow[k];
    a.y = arow[k + 1];
    b.x = bcol[(size_t)k * D_OUT];
    b.y = bcol[(size_t)(k + 1) * D_OUT];
    // (neg_a, A, neg_b, B, c_mod, C, reuse_a, reuse_b)
    c = __builtin_amdgcn_wmma_f32_16x16x4_f32(
        false, a, false, b, (short)0, c, false, false);
  }

  // C/D layout: VGPR i -> row = row0 + i + 8*(lane>=16), col = ncol
  const int rbase = row0 + ((lane >> 4) << 3);
  float* outp = support + (size_t)rbase * D_OUT + ncol;
#pragma unroll
  for (int i = 0; i < 8; ++i)
    outp[(size_t)i * D_OUT] = c[i];
}

// ---------------------------------------------------------------------------
// Kernel 2: row_ptr[r] = lower_bound(adj_rows, r). adj_rows is sorted int64.
// row_ptr has N_NODES+1 entries (row_ptr[N] == N_EDGES automatically).
// ---------------------------------------------------------------------------
__global__ void gcn_build_rowptr(const long long* __restrict__ rows,
                                 int* __restrict__ rowptr,
                                 int n_edges, int n_nodes) {
  const int r = blockIdx.x * blockDim.x + threadIdx.x;
  if (r > n_nodes) return;
  int lo = 0, hi = n_edges;
  const long long key = (long long)r;
  while (lo < hi) {
    const int mid = (lo + hi) >> 1;
    if (rows[mid] < key) lo = mid + 1; else hi = mid;
  }
  rowptr[r] = lo;
}

// ---------------------------------------------------------------------------
// Kernel 3: deterministic segmented SpMM + bias + ReLU.
// One wave32 per node; lane owns features {lane, lane+32}. Each edge pulls a
// 256B coalesced row of support (hot in the 192MB L2: support is 25.6MB).
// ---------------------------------------------------------------------------
__global__ __launch_bounds__(256) void gcn_spmm_relu(
    const float* __restrict__ support,   // [N_NODES, D_OUT]
    const long long* __restrict__ cols,  // [N_EDGES]
    const float* __restrict__ vals,      // [N_EDGES]
    const int* __restrict__ rowptr,      // [N_NODES+1]
    const float* __restrict__ bias,      // [D_OUT]
    float* __restrict__ out,             // [N_NODES, D_OUT]
    int n_nodes) {
  const int lane = threadIdx.x & 31;
  const int node = blockIdx.x * 8 + (threadIdx.x >> 5);
  if (node >= n_nodes) return;

  const int e0 = rowptr[node];
  const int e1 = rowptr[node + 1];

  float acc0 = 0.0f, acc1 = 0.0f;
  for (int e = e0; e < e1; ++e) {
    const float v = vals[e];              // uniform across wave -> scalarized
    const long long cidx = cols[e];
    const float* sp = support + (size_t)cidx * D_OUT;
    acc0 += v * sp[lane];
    acc1 += v * sp[lane + 32];
  }

  acc0 += bias[lane];
  acc1 += bias[lane + 32];
  float* op = out + (size_t)node * D_OUT;
  op[lane]      = acc0 > 0.0f ? acc0 : 0.0f;
  op[lane + 32] = acc1 > 0.0f ? acc1 : 0.0f;
}

// ---------------------------------------------------------------------------
// Launcher. Inputs (setup_inputs order):
//   0: x         float32 [100000*256]
//   1: adj_rows  int64   [1600000]
//   2: adj_cols  int64   [1600000]
//   3: adj_vals  float32 [1600000]
//   4: weight    float32 [256*64]
//   5: bias      float32 [64]
// Output: float32 [100000*64]
// Workspace: support (25.6MB) @ 0, row_ptr (400KB+4) after it.
// ---------------------------------------------------------------------------
extern "C" void kernel_launch(void* const* d_in, const int* in_sizes, int n_in,
                              void* d_out, int out_size, void* d_ws, size_t ws_size,
                              hipStream_t stream) {
  const float*     x        = (const float*)d_in[0];
  const long long* adj_rows = (const long long*)d_in[1];
  const long long* adj_cols = (const long long*)d_in[2];
  const float*     adj_vals = (const float*)d_in[3];
  const float*     weight   = (const float*)d_in[4];
  const float*     bias     = (const float*)d_in[5];
  float*           out      = (float*)d_out;

  const int n_edges = in_sizes[1];

  float* support = (float*)d_ws;
  int*   rowptr  = (int*)((char*)d_ws + (size_t)N_NODES * D_OUT * sizeof(float));

  // 1) Dense GEMM on the WMMA pipe: 6250 M-tiles x (4 waves = 4 N-tiles).
  gcn_gemm_wmma<<<N_NODES / 16, 128, 0, stream>>>(x, weight, support);

  // 2) CSR row pointers from the sorted edge-row list.
  gcn_build_rowptr<<<(N_NODES + 1 + 255) / 256, 256, 0, stream>>>(
      adj_rows, rowptr, n_edges, N_NODES);

  // 3) Deterministic segmented reduction + bias + ReLU (one wave per node).
  gcn_spmm_relu<<<(N_NODES + 7) / 8, 256, 0, stream>>>(
      support, adj_cols, adj_vals, rowptr, bias, out, N_NODES);
}